// VQGlobalProbAvgPool_71829033058532
// MI455X (gfx1250) — compile-verified
//
#include <hip/hip_runtime.h>

#define B_      32
#define L_      2048
#define D_      1024
#define NC_     320
#define NCHUNK  16
#define CHUNK   128   // L_ / NCHUNK
#define PIPE    4     // async pipeline depth

// ---------------- CDNA5 async global->LDS path (guarded) ----------------
#if __has_builtin(__builtin_amdgcn_global_load_async_to_lds_b128)
#define HAVE_ASYNC 1
typedef int v4i_t __attribute__((ext_vector_type(4)));
typedef __attribute__((address_space(1))) v4i_t as1_v4i_t;
typedef __attribute__((address_space(3))) v4i_t as3_v4i_t;
#define ASYNC_COPY16(gsrc, ldst)                                            \
  __builtin_amdgcn_global_load_async_to_lds_b128(                           \
      (as1_v4i_t*)(gsrc), (as3_v4i_t*)(ldst), 0, 0)
#else
#define ASYNC_COPY16(gsrc, ldst)                                            \
  do { *(float4*)(ldst) = *(const float4*)(gsrc); } while (0)
#endif

#ifdef HAVE_ASYNC
#if __has_builtin(__builtin_amdgcn_s_wait_asynccnt)
#define WAIT_ASYNC(n) __builtin_amdgcn_s_wait_asynccnt(n)
#else
#define WAIT_ASYNC(n) asm volatile("s_wait_asynccnt %0" :: "n"(n) : "memory")
#endif
#else
#define WAIT_ASYNC(n) do {} while (0)
#endif

// ---------------- Kernel 3 (placed first for disasm visibility):
// streaming weighted sum via async global->LDS pipeline --------------------
__global__ __launch_bounds__(256)
void k_pool(const float* __restrict__ xfeat, const float* __restrict__ w,
            float* __restrict__ partial) {
  const int b   = blockIdx.x;   // batch
  const int c   = blockIdx.y;   // L-chunk
  const int tid = threadIdx.x;
  const int wv  = tid >> 5;     // wave 0..7, owns d-slice of 128 floats
  const int ln  = tid & 31;

  __shared__ float wsh[CHUNK];
  __shared__ __align__(16) float xb[8][PIPE][128]; // per-wave pipelined rows

  const int l0 = c * CHUNK;
  if (tid < CHUNK) wsh[tid] = w[(size_t)b * L_ + l0 + tid];
  __syncthreads();

  // x = input_feature[:, -1] : layer index N_LAYERS-1 = 1
  const float* src = xfeat
      + (((size_t)(b * 2 + 1)) * L_ + (size_t)l0) * D_
      + wv * 128 + ln * 4;
  float* myslab = &xb[wv][0][0];   // PIPE slots of 128 floats each

  // prologue: fill the pipeline
#pragma unroll
  for (int i = 0; i < PIPE; ++i)
    ASYNC_COPY16(src + (size_t)i * D_, myslab + i * 128 + ln * 4);

  float4 acc = make_float4(0.f, 0.f, 0.f, 0.f);

#define CONSUME_ROW(I)                                                      \
  do {                                                                      \
    asm volatile("" ::: "memory");                                          \
    const float4 v = *(const float4*)(myslab + ((I) & (PIPE - 1)) * 128 + ln * 4); \
    const float ww = wsh[(I)];                                              \
    acc.x = fmaf(ww, v.x, acc.x);                                           \
    acc.y = fmaf(ww, v.y, acc.y);                                           \
    acc.z = fmaf(ww, v.z, acc.z);                                           \
    acc.w = fmaf(ww, v.w, acc.w);                                           \
    asm volatile("" ::: "memory");                                          \
  } while (0)

  // steady state: wait oldest, consume, refill same slot
#pragma unroll 4
  for (int i = 0; i < CHUNK - PIPE; ++i) {
    WAIT_ASYNC(PIPE - 1);
    CONSUME_ROW(i);
    ASYNC_COPY16(src + (size_t)(i + PIPE) * D_,
                 myslab + (i & (PIPE - 1)) * 128 + ln * 4);
  }
  // epilogue: drain with tightening waits
  WAIT_ASYNC(3); CONSUME_ROW(CHUNK - 4);
  WAIT_ASYNC(2); CONSUME_ROW(CHUNK - 3);
  WAIT_ASYNC(1); CONSUME_ROW(CHUNK - 2);
  WAIT_ASYNC(0); CONSUME_ROW(CHUNK - 1);
#undef CONSUME_ROW

  float* outp = partial + ((size_t)(b * NCHUNK + c)) * D_ + wv * 128 + ln * 4;
  *(float4*)outp = acc;
}

// ---------------- Kernel 1: count_x = row sums, count_y = col sums ------
__global__ __launch_bounds__(64)
void k_counts(const float* __restrict__ freqs,
              float* __restrict__ cx, float* __restrict__ cy) {
  const int i = blockIdx.x;     // 0..319
  const int t = threadIdx.x;    // 0..63
  float sr = 0.f, sc = 0.f;
  for (int j = t; j < NC_; j += 64) {
    sr += freqs[i * NC_ + j];   // row i
    sc += freqs[j * NC_ + i];   // col i
  }
  __shared__ float r0[64], r1[64];
  r0[t] = sr; r1[t] = sc;
  __syncthreads();
  for (int s = 32; s > 0; s >>= 1) {
    if (t < s) { r0[t] += r0[t + s]; r1[t] += r1[t + s]; }
    __syncthreads();
  }
  if (t == 0) { cx[i] = r0[0]; cy[i] = r1[0]; }
}

// ---------------- Kernel 2: normalized weights w[b,l] -------------------
__global__ __launch_bounds__(256)
void k_weights(const float* __restrict__ cx, const float* __restrict__ cy,
               const int* __restrict__ lens, const int* __restrict__ vq,
               float* __restrict__ w) {
  const int b = blockIdx.x;
  const int t = threadIdx.x;
  const int len = lens[b];
  float p[8];
  float s = 0.f;
#pragma unroll
  for (int k = 0; k < 8; ++k) {
    const int l  = t + k * 256;
    const int i0 = vq[((size_t)b * L_ + l) * 2 + 0];
    const int i1 = vq[((size_t)b * L_ + l) * 2 + 1];
    const float pr = (l < len) ? (1.0f / (cx[i0] + cy[i1])) : 0.0f;
    p[k] = pr;
    s += pr;
  }
  __shared__ float red[256];
  red[t] = s;
  __syncthreads();
  for (int st = 128; st > 0; st >>= 1) {
    if (t < st) red[t] += red[t + st];
    __syncthreads();
  }
  const float inv = 1.0f / red[0];
#pragma unroll
  for (int k = 0; k < 8; ++k)
    w[(size_t)b * L_ + t + k * 256] = p[k] * inv;
}

// ---------------- Kernel 4: fold L-chunk partials -----------------------
__global__ __launch_bounds__(256)
void k_reduce(const float* __restrict__ partial, float* __restrict__ out) {
  const int idx = blockIdx.x * 256 + threadIdx.x;  // 0..B_*D_-1
  const int b = idx >> 10;
  const int d = idx & (D_ - 1);
  float s = 0.f;
#pragma unroll
  for (int c = 0; c < NCHUNK; ++c)
    s += partial[((size_t)(b * NCHUNK + c)) * D_ + d];
  out[idx] = s;
}

// ---------------- Host-side launch --------------------------------------
extern "C" void kernel_launch(void* const* d_in, const int* in_sizes, int n_in,
                              void* d_out, int out_size, void* d_ws, size_t ws_size,
                              hipStream_t stream) {
  const float* input_feature = (const float*)d_in[0]; // (B,2,L,D) f32
  const float* freqs         = (const float*)d_in[1]; // (320,320) f32
  const int*   input_lengths = (const int*)  d_in[2]; // (B,) i32
  const int*   vq_indices    = (const int*)  d_in[3]; // (B,L,2) i32
  float*       out           = (float*)d_out;         // (B,D) f32

  float* ws      = (float*)d_ws;
  float* cx      = ws;                       // 320
  float* cy      = ws + NC_;                 // 320
  float* w       = ws + 2 * NC_;             // B_*L_ = 65536
  float* partial = w + (size_t)B_ * L_;      // B_*NCHUNK*D_ = 524288

  k_counts <<<NC_,           64, 0, stream>>>(freqs, cx, cy);
  k_weights<<<B_,           256, 0, stream>>>(cx, cy, input_lengths, vq_indices, w);
  k_pool   <<<dim3(B_, NCHUNK), 256, 0, stream>>>(input_feature, w, partial);
  k_reduce <<<(B_ * D_) / 256, 256, 0, stream>>>(partial, out);
}